// DynamicPooler_58540404244695
// MI455X (gfx1250) — compile-verified
//
#include <hip/hip_runtime.h>
#include <math.h>

typedef float v2f __attribute__((ext_vector_type(2)));
typedef float v8f __attribute__((ext_vector_type(8)));

// Problem constants (from reference setup_inputs)
constexpr int L   = 2048;
constexpr int Bsz = 4;
constexpr int QH  = 32;
constexpr int KVH = 8;
constexpr int Dch = 128;
constexpr int G   = 4;       // QH / KVH
constexpr int N   = 32;      // Bsz * KVH
constexpr int WS  = 12;
constexpr float INIT_VAL   = 2.0f;
constexpr float MASK_CONST = -50000.0f;
constexpr float EPS_F      = 1.1920929e-07f;

constexpr int TT   = 32;            // t-tile per block
constexpr int HALO = WS - 1;        // 11
constexpr int ROWS = TT + HALO + 1; // 44 rows staged (incl. t+1 row for mask / K-pad row)
constexpr int XS   = 272;           // LDS row stride in floats (256 data + 16 pad -> bank shift)

// Dynamic-LDS layout (floats); no static __shared__ so dynamic base offset == 0
constexpr int OFF_X    = 0;
constexpr int OFF_W    = OFF_X + ROWS * XS;
constexpr int OFF_DEC  = OFF_W + ROWS;
constexpr int OFF_RCP  = OFF_DEC + ROWS;
constexpr int OFF_DMAX = OFF_RCP + TT;
constexpr int SMEM_FLOATS = OFF_DMAX + TT;   // 12120 floats = 48480 B

// q_out: zero last channel, pure float4 streaming copy
__global__ __launch_bounds__(256) void qzero_copy(const float* __restrict__ q,
                                                  float* __restrict__ out, int n4) {
  int i = blockIdx.x * blockDim.x + threadIdx.x;
  if (i < n4) {
    float4 val = ((const float4*)q)[i];
    if ((i & 31) == 31) val.w = 0.0f;   // channels 124..127 -> zero ch 127
    ((float4*)out)[i] = val;
  }
}

__global__ __launch_bounds__(128) void pool_wmma(const float* __restrict__ q,
                                                 const float* __restrict__ k,
                                                 const float* __restrict__ v,
                                                 float* __restrict__ out) {
  extern __shared__ __align__(16) float smem[];
  float* const x_lds   = smem + OFF_X;
  float* const w_lds   = smem + OFF_W;
  float* const dec_lds = smem + OFF_DEC;
  float* const rcp_lds = smem + OFF_RCP;
  int*   const dmax_lds = (int*)(smem + OFF_DMAX);

  const int tid    = threadIdx.x;
  const int lane   = tid & 31;
  const int laneHi = (lane >> 4) & 1;
  const int M      = lane & 15;
  const int wv     = tid >> 5;        // wave id 0..3

  const int n  = blockIdx.x % N;
  const int t0 = (blockIdx.x / N) * TT;
  const int b  = n / KVH;
  const int h  = n % KVH;

  // ---- stage k||v tile rows [t0-11 .. t0+32] into LDS via async global->LDS (B128) ----
  // 44 rows x 64 float4 (k -> ch 0..127, v -> ch 128..255); OOB rows zero-filled via ds_store.
  for (int i = tid; i < ROWS * 64; i += 128) {
    const int row = i >> 6;
    const int qq  = i & 63;           // 0..31 -> k float4, 32..63 -> v float4
    const int gr  = t0 - HALO + row;
    const unsigned ldsoff = (unsigned)(row * XS + qq * 4) * 4u;   // byte offset in LDS
    if (gr >= 0 && gr < L) {
      const size_t rowbase = (size_t)(gr * N + n) * Dch;
      const float* src = (qq < 32) ? (k + rowbase + qq * 4)
                                   : (v + rowbase + (qq - 32) * 4);
      const unsigned long long gaddr = (unsigned long long)(uintptr_t)src;
      asm volatile("global_load_async_to_lds_b128 %0, %1, off"
                   :: "v"(ldsoff), "v"(gaddr) : "memory");
    } else {
      *(float4*)&x_lds[row * XS + qq * 4] = make_float4(0.f, 0.f, 0.f, 0.f);
    }
  }

  // ---- decisions + token weights per staged row ----
  for (int r = tid; r < ROWS; r += 128) {
    const int gr = t0 - HALO + r;
    float dec = 0.f, wt = 1.f;
    if (gr >= 0 && gr < L) {
      dec = (k[(size_t)(gr * N + n) * Dch + (Dch - 1)] > INIT_VAL) ? 1.f : 0.f;
      float s = 0.f;
#pragma unroll
      for (int g = 0; g < G; ++g)
        s += q[(((size_t)gr * Bsz + b) * QH + (h * G + g)) * Dch + (Dch - 1)];
      s = s * 0.25f + INIT_VAL;
      wt = 1.f / (1.f + expf(-s)) + EPS_F;
    }
    dec_lds[r] = dec;
    w_lds[r]   = wt;
  }
  // drain the async tile loads, then make all LDS writes visible block-wide
  asm volatile("s_wait_asynccnt 0" ::: "memory");
  __syncthreads();

  // ---- per-t window depth (run of consecutive 1-decisions) + normalization, mask out ----
  float* const mask_out = out + (size_t)3 * L * Bsz * QH * Dch;
  if (tid < TT) {
    const int tg = t0 + tid;
    int r = 0;
    while (r < HALO && dec_lds[HALO + tid - r] > 0.5f) r++;
    const int dmax = (r < tg) ? r : tg;         // min(runlen, 11, t)
    float c = 0.f;
    for (int d = 0; d <= dmax; ++d) c += w_lds[HALO + tid - d];
    dmax_lds[tid] = dmax;
    rcp_lds[tid]  = 1.f / c;
    const float m = (tg == L - 1) ? 0.f : dec_lds[HALO + tid + 1] * MASK_CONST;
#pragma unroll
    for (int g = 0; g < G; ++g)
      mask_out[((size_t)tg * Bsz + b) * QH + h * G + g] = m;
  }
  __syncthreads();

  float* const k2_out = out + (size_t)L * Bsz * QH * Dch;
  float* const v2_out = k2_out + (size_t)L * Bsz * QH * Dch;

  // ---- pooled(16x256) = A(16x28) @ X(28x256) via V_WMMA_F32_16X16X4_F32 ----
  for (int rb = 0; rb < 2; ++rb) {
    // A fragment per k-step: ISA 16x4 f32 A layout -> lane M holds K = j + 2*laneHi
    const int dmaxM = dmax_lds[rb * 16 + M];
    v2f A[7];
#pragma unroll
    for (int s = 0; s < 7; ++s) {
      v2f a;
#pragma unroll
      for (int j = 0; j < 2; ++j) {
        const int kabs = 4 * s + j + 2 * laneHi;   // 0..27
        const int d    = M + HALO - kabs;          // lag
        float c = 0.f;
        if (d >= 0 && d <= dmaxM) c = w_lds[rb * 16 + kabs];
        a[j] = c;
      }
      A[s] = a;
    }
#pragma unroll
    for (int ctl = 0; ctl < 4; ++ctl) {
      const int ct = wv * 4 + ctl;                 // 16-ch column tile, 0..15
      v8f C = {0.f, 0.f, 0.f, 0.f, 0.f, 0.f, 0.f, 0.f};
#pragma unroll
      for (int s = 0; s < 7; ++s) {
        v2f Bf;
#pragma unroll
        for (int j = 0; j < 2; ++j) {
          const int kk = 4 * s + j + 2 * laneHi;   // 0..27; row 43 staged (zeros/unused)
          Bf[j] = x_lds[(rb * 16 + kk) * XS + ct * 16 + M];
        }
        C = __builtin_amdgcn_wmma_f32_16x16x4_f32(false, A[s], false, Bf,
                                                  (short)0, C, false, false);
      }
      // scale by 1/c and broadcast-store across the G=4 grouped query heads
      const bool isK  = (ct < 8);
      const int  ch   = isK ? (ct * 16 + M) : (ct * 16 + M - 128);
      const bool zK   = isK && (ch == Dch - 1);    // k2 last channel zeroed
      float* const base = isK ? k2_out : v2_out;
#pragma unroll
      for (int j = 0; j < 8; ++j) {
        const int tl  = rb * 16 + j + 8 * laneHi;  // C layout: vgpr j -> M=j (+8 hi lanes)
        const float val = zK ? 0.f : C[j] * rcp_lds[tl];
        const size_t o = (((size_t)(t0 + tl) * Bsz + b) * QH + h * G) * Dch + ch;
        base[o]            = val;
        base[o + Dch]      = val;
        base[o + 2 * Dch]  = val;
        base[o + 3 * Dch]  = val;
      }
    }
  }
}

extern "C" void kernel_launch(void* const* d_in, const int* in_sizes, int n_in,
                              void* d_out, int out_size, void* d_ws, size_t ws_size,
                              hipStream_t stream) {
  const float* q = (const float*)d_in[0];
  const float* k = (const float*)d_in[1];
  const float* v = (const float*)d_in[2];
  float* out = (float*)d_out;

  const int n4 = L * Bsz * QH * Dch / 4;   // 8388608 float4
  qzero_copy<<<(n4 + 255) / 256, 256, 0, stream>>>(q, out, n4);
  pool_wmma<<<(L / TT) * N, 128, SMEM_FLOATS * sizeof(float), stream>>>(q, k, v, out);
}